// ODE_vp_func_14637248544946
// MI455X (gfx1250) — compile-verified
//
#include <hip/hip_runtime.h>
#include <math.h>

#define BLOCK_SZ 256

// Exact pointer types for the gfx1250 async-LDS builtin:
//   param0: v4i (int vector_size(16)) in global addrspace(1)
//   param1: v4i in LDS addrspace(3)
//   param2/3: imm offset, imm cpol
typedef int v4i __attribute__((vector_size(16)));
typedef v4i __attribute__((address_space(1)))* global_v4i_ptr;
typedef v4i __attribute__((address_space(3)))* lds_v4i_ptr;

// ---------- CDNA5 async global->LDS copy (16B per active lane) ----------
__device__ __forceinline__ void async_copy_b128(float* lds_dst, const float* g_src) {
#if __has_builtin(__builtin_amdgcn_global_load_async_to_lds_b128)
    __builtin_amdgcn_global_load_async_to_lds_b128(
        (global_v4i_ptr)(__attribute__((address_space(1))) void*)(void*)g_src,
        (lds_v4i_ptr)(__attribute__((address_space(3))) void*)(void*)lds_dst,
        /*offset=*/0, /*cpol=*/0);
#else
    unsigned loff = (unsigned)(size_t)(__attribute__((address_space(3))) float*)lds_dst;
    asm volatile("global_load_async_to_lds_b128 %0, %1, off"
                 :: "v"(loff), "v"(g_src)
                 : "memory");
#endif
}

__device__ __forceinline__ void wait_asynccnt0() {
#if __has_builtin(__builtin_amdgcn_s_wait_asynccnt)
    __builtin_amdgcn_s_wait_asynccnt(0);
#else
    asm volatile("s_wait_asynccnt 0x0" ::: "memory");
#endif
}

// Memory-bound element-wise ODE RHS. One thread per batch element.
// Full 256-element tiles stage the stride-12 u rows through LDS via
// async b128 loads (coalesced 16B/lane) drained before the direct quat b128 loads.
__global__ __launch_bounds__(BLOCK_SZ) void ode_vp_kernel(
    const float* __restrict__ tp,   // t scalar
    const float* __restrict__ y,    // (n, 6)
    const float* __restrict__ u,    // (T, n, 3)
    const float* __restrict__ quat, // (T, n, 4)
    const float* __restrict__ dtp,  // Dt scalar
    float* __restrict__ out,        // (n, 6)
    int n, int T)
{
    __shared__ float su0[3 * BLOCK_SZ];   // 3 KB
    __shared__ float su1[3 * BLOCK_SZ];   // 3 KB

    const float t  = tp[0];
    const float Dt = dtp[0];

    const int  ii       = (int)floorf(t / Dt);
    const bool in_range = ii < (T - 1);
    int i0 = ii;
    if (i0 < 0) i0 = 0;
    if (i0 > T - 1) i0 = T - 1;
    int i1 = ii + 1;
    if (i1 < 0) i1 = 0;
    if (i1 > T - 1) i1 = T - 1;

    const float tau    = t - (float)i0 * Dt;
    const float lerp_s = in_range ? (tau / Dt) : 0.0f;

    const size_t u0base = (size_t)i0 * 3u * (size_t)n;
    const size_t u1base = (size_t)i1 * 3u * (size_t)n;
    const size_t q0base = (size_t)i0 * 4u * (size_t)n;
    const size_t q1base = (size_t)i1 * 4u * (size_t)n;

    const int tid  = (int)threadIdx.x;
    const int tile = (int)blockIdx.x * BLOCK_SZ;
    const int b    = tile + tid;
    const bool full_tile = (tile + BLOCK_SZ) <= n;

    float vx, vy, vz;

    if (full_tile) {
        // ---- async-stage the two u tiles: 3*256 floats = 3072B = 192 lanes x 16B
        if (tid < (3 * BLOCK_SZ) / 4) {
            const float* g0 = u + u0base + 3u * (size_t)tile + 4u * (size_t)tid;
            const float* g1 = u + u1base + 3u * (size_t)tile + 4u * (size_t)tid;
            async_copy_b128(su0 + 4 * tid, g0);
            async_copy_b128(su1 + 4 * tid, g1);
        }
        wait_asynccnt0();     // drain this wave's async ops
        __syncthreads();      // make all waves' LDS writes visible

        vx = su0[3 * tid + 0];
        vy = su0[3 * tid + 1];
        vz = su0[3 * tid + 2];
        vx += lerp_s * (su1[3 * tid + 0] - vx);
        vy += lerp_s * (su1[3 * tid + 1] - vy);
        vz += lerp_s * (su1[3 * tid + 2] - vz);
    } else {
        if (b >= n) return;
        const float* u0p = u + u0base + 3u * (size_t)b;
        const float* u1p = u + u1base + 3u * (size_t)b;
        vx = u0p[0]; vy = u0p[1]; vz = u0p[2];
        vx += lerp_s * (u1p[0] - vx);
        vy += lerp_s * (u1p[1] - vy);
        vz += lerp_s * (u1p[2] - vz);
    }

    // ---- slerp of quat (direct b128 loads, already fully coalesced) ----
    // VALU-reduced form: with theta = acos(d),
    //   sin(theta)        = sqrt(1 - d^2)
    //   w1 = sin(tau*theta)/sin(theta)
    //   w0 = sin((1-tau)*theta)/sin(theta) = cos(tau*theta) - d*w1
    const float4 q0 = *(const float4*)(quat + q0base + 4u * (size_t)b);
    const float4 q1 = *(const float4*)(quat + q1base + 4u * (size_t)b);
    float d = q0.x * q1.x + q0.y * q1.y + q0.z * q1.z + q0.w * q1.w;
    const float sgn = (d < 0.0f) ? -1.0f : 1.0f;   // flip q1 if dot < 0
    d = fabsf(d);
    d = fminf(d, 1.0f);
    const float dc    = fminf(d, 1.0f - 1e-7f);
    const float theta = acosf(dc);
    const float st    = sqrtf(fmaxf(1.0f - dc * dc, 0.0f));  // sin(theta) >= 0 on [0, pi]
    float sp, cp;
    sincosf(tau * theta, &sp, &cp);                 // one shared argument reduction
    const bool  smallb = st < 1e-6f;
    float w1 = smallb ? tau : (sp / st);
    float w0 = smallb ? (1.0f - tau) : (cp - dc * w1);
    w1 *= sgn;  // fold the q1 sign flip into w1

    float qw, qx, qy, qz;
    if (in_range) {
        qw = w0 * q0.x + w1 * q1.x;
        qx = w0 * q0.y + w1 * q1.y;
        qy = w0 * q0.z + w1 * q1.z;
        qz = w0 * q0.w + w1 * q1.w;
    } else {
        qw = q0.x; qx = q0.y; qy = q0.z; qz = q0.w;
    }

    // ---- rotate v by quaternion: Ra = v + w*t2 + qv x t2, t2 = 2*(qv x v) ----
    const float tx = 2.0f * (qy * vz - qz * vy);
    const float ty = 2.0f * (qz * vx - qx * vz);
    const float tz = 2.0f * (qx * vy - qy * vx);
    const float rx = vx + qw * tx + (qy * tz - qz * ty);
    const float ry = vy + qw * ty + (qz * tx - qx * tz);
    const float rz = vz + qw * tz + (qx * ty - qy * tx) - 9.81f;  // + G.z

    // ---- y passthrough + packed stores (three aligned b64 stores) ----
    const float* yp = y + 6u * (size_t)b;
    const float2 y01 = *(const float2*)yp;
    const float  y2  = yp[2];

    float* op = out + 6u * (size_t)b;
    *(float2*)(op + 0) = make_float2(rx, ry);
    *(float2*)(op + 2) = make_float2(rz, y01.x);
    *(float2*)(op + 4) = make_float2(y01.y, y2);
}

extern "C" void kernel_launch(void* const* d_in, const int* in_sizes, int n_in,
                              void* d_out, int out_size, void* d_ws, size_t ws_size,
                              hipStream_t stream) {
    (void)n_in; (void)out_size; (void)d_ws; (void)ws_size;
    const float* tp   = (const float*)d_in[0];  // t (1)
    const float* y    = (const float*)d_in[1];  // (n,6)
    const float* u    = (const float*)d_in[2];  // (T,n,3)
    const float* quat = (const float*)d_in[3];  // (T,n,4)
    const float* dtp  = (const float*)d_in[4];  // Dt (1)
    float* out = (float*)d_out;

    const int n = in_sizes[1] / 6;
    const int T = (int)((long long)in_sizes[2] / (3LL * (long long)n));

    int grid = (n + BLOCK_SZ - 1) / BLOCK_SZ;
    if (grid < 1) grid = 1;

    ode_vp_kernel<<<grid, BLOCK_SZ, 0, stream>>>(tp, y, u, quat, dtp, out, n, T);
}